// RecurrentLinearAttentionPPLM_13984413515886
// MI455X (gfx1250) — compile-verified
//
#include <hip/hip_runtime.h>
#include <math.h>
#include <stdint.h>

typedef __attribute__((ext_vector_type(2))) float v2f;
typedef __attribute__((ext_vector_type(8))) float v8f;

#define NH    512          // N*H = 32*16
#define Dk    64           // feature dim D
#define Mv    64           // value dim M
#define Tt    2048         // history length T
#define TC    2049         // T+1
#define HALF  1024         // T split across 2 wave-groups
#define STEPS (HALF / 4)   // K-steps of 4 per wave (=256, even)
#define EPSF  1e-6f

__device__ __forceinline__ float elu1(float x) {
    // elu(x)+1 : strictly positive feature map
    return x > 0.0f ? (x + 1.0f) : __expf(x);
}

#define WMMA_F32(A, B, C) \
    __builtin_amdgcn_wmma_f32_16x16x4_f32(false, (A), false, (B), (short)0, (C), false, false)

// One workgroup (8 waves) per head.
// wave w: g = w>>2 selects T-half; dt = w&3 selects d-rows [16dt,16dt+16).
__global__ __launch_bounds__(256) void rla_main(
    const float* __restrict__ query,
    const float* __restrict__ key,
    const float* __restrict__ value,
    const float* __restrict__ k_hist,
    const float* __restrict__ v_hist,
    float* __restrict__ outV,   // [NH, M]
    float* __restrict__ outK,   // [NH, D, T+1]
    float* __restrict__ outVc)  // [NH, M, T+1]
{
    const int nh  = blockIdx.x;
    const int tid = threadIdx.x;
    const int w   = tid >> 5;
    const int g   = w >> 2;        // T-half index (0/1)
    const int dt  = w & 3;         // d-tile index (0..3)
    const int l   = tid & 31;
    const int hi  = l >> 4;        // lane group (0/1) -> K slots 2*hi..2*hi+1
    const int lo  = l & 15;

    __shared__ float s_col[Mv];    // sum_d Q[d]*Si[d,m]
    __shared__ float s_den;        // Q . Zi
    if (tid < Mv) s_col[tid] = 0.0f;
    if (tid == 0) s_den = 0.0f;
    __syncthreads();

    // Uniform (scalar) head base pointers; per-lane 32-bit element offsets.
    const float* kb  = k_hist + (size_t)nh * Dk * Tt;
    const float* vb  = v_hist + (size_t)nh * Dk * Tt;
    float*       okb = outK   + (size_t)nh * Dk * TC;
    float*       ovb = outVc  + (size_t)nh * Dk * TC;

    const uint32_t arow = (uint32_t)(dt * 16 + lo); // A row (d); also own V_cat row
    const uint32_t jb   = (uint32_t)(g * HALF + 2 * hi);
    uint32_t aoff = arow * Tt + jb;                 // k_hist / own v_hist row offset
    uint32_t boff = (uint32_t)lo * Tt + jb;         // B tile 0 (tiles add 16*Tt as imm)
    uint32_t ooff = arow * TC + jb;                 // K_cat / V_cat store offset

    v8f acc[4];
    #pragma unroll
    for (int t = 0; t < 4; ++t) acc[t] = (v8f){0.f,0.f,0.f,0.f,0.f,0.f,0.f,0.f};
    float zi = 0.0f;

    // process one K-step: fused copies + zi + 4 WMMAs
    auto PROC = [&](v2f a, v2f b0, v2f b1, v2f b2, v2f b3, v2f vo, uint32_t oo) {
        okb[oo]     = a.x;
        okb[oo + 1] = a.y;
        ovb[oo]     = vo.x;      // own m-quarter copy (row arow), no select needed
        ovb[oo + 1] = vo.y;
        zi += a.x + a.y;
        acc[0] = WMMA_F32(a, b0, acc[0]);
        acc[1] = WMMA_F32(a, b1, acc[1]);
        acc[2] = WMMA_F32(a, b2, acc[2]);
        acc[3] = WMMA_F32(a, b3, acc[3]);
    };

    // ---- 2-deep software pipeline, unrolled 2x ----
    v2f a0  = *(const v2f*)(kb + aoff);
    v2f b00 = *(const v2f*)(vb + boff);
    v2f b01 = *(const v2f*)(vb + boff + 16 * Tt);
    v2f b02 = *(const v2f*)(vb + boff + 32 * Tt);
    v2f b03 = *(const v2f*)(vb + boff + 48 * Tt);
    v2f vo0 = *(const v2f*)(vb + aoff);

    v2f a1  = *(const v2f*)(kb + aoff + 4);
    v2f b10 = *(const v2f*)(vb + boff + 4);
    v2f b11 = *(const v2f*)(vb + boff + 4 + 16 * Tt);
    v2f b12 = *(const v2f*)(vb + boff + 4 + 32 * Tt);
    v2f b13 = *(const v2f*)(vb + boff + 4 + 48 * Tt);
    v2f vo1 = *(const v2f*)(vb + aoff + 4);

    for (int it = 0; it < STEPS - 2; it += 2) {
        // consume set 0 (step it), then refill it with step it+2
        PROC(a0, b00, b01, b02, b03, vo0, ooff);
        a0  = *(const v2f*)(kb + aoff + 8);
        b00 = *(const v2f*)(vb + boff + 8);
        b01 = *(const v2f*)(vb + boff + 8 + 16 * Tt);
        b02 = *(const v2f*)(vb + boff + 8 + 32 * Tt);
        b03 = *(const v2f*)(vb + boff + 8 + 48 * Tt);
        vo0 = *(const v2f*)(vb + aoff + 8);

        // consume set 1 (step it+1), then refill it with step it+3
        PROC(a1, b10, b11, b12, b13, vo1, ooff + 4);
        a1  = *(const v2f*)(kb + aoff + 12);
        b10 = *(const v2f*)(vb + boff + 12);
        b11 = *(const v2f*)(vb + boff + 12 + 16 * Tt);
        b12 = *(const v2f*)(vb + boff + 12 + 32 * Tt);
        b13 = *(const v2f*)(vb + boff + 12 + 48 * Tt);
        vo1 = *(const v2f*)(vb + aoff + 12);

        aoff += 8; boff += 8; ooff += 8;
    }
    // tail: last two K-steps
    PROC(a0, b00, b01, b02, b03, vo0, ooff);
    PROC(a1, b10, b11, b12, b13, vo1, ooff + 4);

    // ---- appended column j = T: rank-1 update in C-fragment layout (group 0 only)
    // C layout: vgpr r, lane -> Si[16*dt + r + 8*hi][16t + lo]
    if (g == 0) {
        float vv[4];
        #pragma unroll
        for (int t = 0; t < 4; ++t) vv[t] = value[nh * Mv + t * 16 + lo];
        #pragma unroll
        for (int r = 0; r < 8; ++r) {
            const float kf = elu1(key[nh * Dk + dt * 16 + r + 8 * hi]);
            #pragma unroll
            for (int t = 0; t < 4; ++t)
                acc[t][r] += kf * vv[t];
        }
    }

    // ---- denominator: Q . Zi  (appended K column counted once: g==0, hi==0)
    {
        float zr = zi;
        if (g == 0 && hi == 0) zr += elu1(key[nh * Dk + arow]);
        const float qa = elu1(query[nh * Dk + arow]);
        atomicAdd(&s_den, qa * zr);                       // ds_add_f32
    }

    // ---- numerator: per-lane partial of sum_d Q[d]*Si[d,m]
    float qc[8];
    #pragma unroll
    for (int r = 0; r < 8; ++r)
        qc[r] = elu1(query[nh * Dk + dt * 16 + r + 8 * hi]);
    #pragma unroll
    for (int t = 0; t < 4; ++t) {
        float cp = 0.0f;
        #pragma unroll
        for (int r = 0; r < 8; ++r) cp += qc[r] * acc[t][r];
        atomicAdd(&s_col[t * 16 + lo], cp);               // ds_add_f32
    }

    // ---- appended column j = T of K_cat / V_cat for this head (fused; no 2nd kernel)
    // Overlaps the LDS reduction; one store pair per d-row, done by tid < 64.
    if (tid < Dk) {
        const int i = nh * Dk + tid;
        okb[(size_t)tid * TC + Tt] = elu1(key[i]);
        ovb[(size_t)tid * TC + Tt] = value[i];
    }

    __syncthreads();

    if (tid < Mv) {
        const float Z = 1.0f / (s_den + EPSF);
        outV[(size_t)nh * Mv + tid] = Z * s_col[tid];
    }
}

extern "C" void kernel_launch(void* const* d_in, const int* in_sizes, int n_in,
                              void* d_out, int out_size, void* d_ws, size_t ws_size,
                              hipStream_t stream) {
    const float* query  = (const float*)d_in[0];
    const float* key    = (const float*)d_in[1];
    const float* value  = (const float*)d_in[2];
    const float* k_hist = (const float*)d_in[3];
    const float* v_hist = (const float*)d_in[4];

    float* outV  = (float*)d_out;                        // [NH, M]
    float* outK  = outV + (size_t)NH * Mv;               // [NH, D, T+1]
    float* outVc = outK + (size_t)NH * Dk * TC;          // [NH, M, T+1]

    rla_main<<<NH, 256, 0, stream>>>(query, key, value, k_hist, v_hist,
                                     outV, outK, outVc);
}